// RagWhiteBox_14319420965178
// MI455X (gfx1250) — compile-verified
//
#include <hip/hip_runtime.h>
#include <math.h>
#include <stdint.h>

// ---------------------------------------------------------------------------
// RAG white-box pipeline on MI455X (gfx1250), fp32 end-to-end via
// V_WMMA_F32_16X16X4_F32, with GLOBAL_LOAD_ASYNC_TO_LDS_B128 double-buffered
// staging (ASYNCcnt-tracked, no ds_store / no VGPR staging).
// wave32; 256-thread blocks = 8 waves; 32x32 C tile per wave (4 accumulators).
// ---------------------------------------------------------------------------

typedef __attribute__((ext_vector_type(2))) float v2f;
typedef __attribute__((ext_vector_type(8))) float v8f;

#define B_ROWS   1024
#define SEQ_T    200
#define DIM      768      // N_EMBD
#define HID      3072     // 4*N_EMBD
#define NVOCAB   32000
#define CAP      100000
#define NCHUNK   512      // index columns per block (4 subtiles of 128)
#define NCHUNKS  196      // ceil(100000 / 512)
#define PADA     20       // padded row of a [*][16] k-major tile: 80B, 16B-aligned

// Async global->LDS copy of 16 bytes per lane; tracked by ASYNCcnt.
__device__ __forceinline__ void async_copy_b128(uint32_t lds_addr, const float* gsrc) {
  asm volatile("global_load_async_to_lds_b128 %0, %1, off"
               :: "v"(lds_addr), "v"(gsrc) : "memory");
}
// Wait until <=3 async ops outstanding (previous 3-op stage complete; async
// loads complete in order) / until all complete.
__device__ __forceinline__ void async_wait_le3() {
  asm volatile("s_wait_asynccnt 0x3" ::: "memory");
}
__device__ __forceinline__ void async_wait_0() {
  asm volatile("s_wait_asynccnt 0x0" ::: "memory");
}

__device__ __forceinline__ uint32_t lds_addr_of(const void* p) {
  // Generic (flat) pointers to LDS carry the wave-relative LDS byte offset in
  // their low 32 bits.
  return (uint32_t)(size_t)p;
}

__device__ __forceinline__ void top3_insert(float s, int n,
                                            float& v0, float& v1, float& v2,
                                            int& i0, int& i1, int& i2) {
  if (s > v0)      { v2 = v1; i2 = i1; v1 = v0; i1 = i0; v0 = s; i0 = n; }
  else if (s > v1) { v2 = v1; i2 = i1; v1 = s;  i1 = n; }
  else if (s > v2) { v2 = s;  i2 = n; }
}

// ---------------------------------------------------------------------------
// 1) query = mean_t embed[ids[b,t]]; qn = query / max(||query||, 1e-12)
// ---------------------------------------------------------------------------
__global__ __launch_bounds__(256) void query_kernel(
    const int* __restrict__ ids, const float* __restrict__ embed,
    float* __restrict__ query, float* __restrict__ qn) {
  const int b = blockIdx.x;
  const int tid = threadIdx.x;
  float a0 = 0.f, a1 = 0.f, a2 = 0.f;
  const int* row_ids = ids + (size_t)b * SEQ_T;
  for (int t = 0; t < SEQ_T; ++t) {
    const float* e = embed + (size_t)row_ids[t] * DIM;
    a0 += e[tid];
    a1 += e[tid + 256];
    a2 += e[tid + 512];
  }
  const float inv_t = 1.0f / (float)SEQ_T;
  a0 *= inv_t; a1 *= inv_t; a2 *= inv_t;
  float* qrow = query + (size_t)b * DIM;
  qrow[tid] = a0; qrow[tid + 256] = a1; qrow[tid + 512] = a2;

  __shared__ float red[256];
  red[tid] = a0 * a0 + a1 * a1 + a2 * a2;
  __syncthreads();
  for (int s = 128; s > 0; s >>= 1) {
    if (tid < s) red[tid] += red[tid + s];
    __syncthreads();
  }
  const float inv = 1.0f / fmaxf(sqrtf(red[0]), 1e-12f);
  float* qnrow = qn + (size_t)b * DIM;
  qnrow[tid] = a0 * inv; qnrow[tid + 256] = a1 * inv; qnrow[tid + 512] = a2 * inv;
}

// ---------------------------------------------------------------------------
// 2) scores = qn @ index^T with fused per-chunk top-3.
//    Grid: x = mtile (16, fast => 16 blocks share an index chunk in L2),
//          y = chunk (196).  Block tile 64 x 512 (4 subtiles of 64x128).
//    Wave w: 32x32 C tile at ((w>>2)*32, (w&3)*32).
// ---------------------------------------------------------------------------
__global__ __launch_bounds__(256) void scores_topk_kernel(
    const float* __restrict__ qn, const float* __restrict__ index,
    float* __restrict__ pv, int* __restrict__ pi) {
  const int mtile = blockIdx.x;
  const int chunk = blockIdx.y;
  const int tid  = threadIdx.x;
  const int wave = tid >> 5;
  const int lane = tid & 31;
  const int lr   = lane & 15;
  const int khalf = (lane >> 4) << 1;   // K offset 0/2 (ISA 16x4 A/B layout)
  const int hi8  = (lane >> 4) << 3;    // C row offset 0/8
  const int w_m  = (wave >> 2) << 5;    // 0,32
  const int w_n  = (wave & 3) << 5;     // 0,32,64,96
  const int m0   = mtile * 64;

  __shared__ float As[2][64][PADA];     // qn tile    [m][k]
  __shared__ float Bs[2][128][PADA];    // index tile [n][k] (natural layout)
  __shared__ float Cs[64][132];         // score subtile staging

  float v0 = -__builtin_inff(), v1 = -__builtin_inff(), v2 = -__builtin_inff();
  int   i0 = 0, i1 = 0, i2 = 0;

  const int arow  = tid >> 2;           // 0..63
  const int ak4   = (tid & 3) << 2;     // 0,4,8,12
  const int brow0 = tid >> 2;           // 0..63
  const int brow1 = 64 + (tid >> 2);    // 64..127
  const float* aG = qn + (size_t)(m0 + arow) * DIM + ak4;

  for (int sub = 0; sub < 4; ++sub) {
    const int n0 = chunk * NCHUNK + sub * 128;
    // Clamp rows (uniform EXEC in the hot loop); masked in the top-3 scan.
    const int nr0 = (n0 + brow0 < CAP) ? (n0 + brow0) : (CAP - 1);
    const int nr1 = (n0 + brow1 < CAP) ? (n0 + brow1) : (CAP - 1);
    const float* bG0 = index + (size_t)nr0 * DIM + ak4;
    const float* bG1 = index + (size_t)nr1 * DIM + ak4;

    // Prologue: stage kc=0 into buffer 0 (3 async instrs per wave).
    async_copy_b128(lds_addr_of(&As[0][arow][ak4]),  aG);
    async_copy_b128(lds_addr_of(&Bs[0][brow0][ak4]), bG0);
    async_copy_b128(lds_addr_of(&Bs[0][brow1][ak4]), bG1);

    v8f c00 = {}, c01 = {}, c10 = {}, c11 = {};
    int cur = 0;
    for (int kc = 0; kc < DIM; kc += 16) {
      if (kc + 16 < DIM) {            // prefetch next stage into other buffer
        const int nb = cur ^ 1;
        async_copy_b128(lds_addr_of(&As[nb][arow][ak4]),  aG  + kc + 16);
        async_copy_b128(lds_addr_of(&Bs[nb][brow0][ak4]), bG0 + kc + 16);
        async_copy_b128(lds_addr_of(&Bs[nb][brow1][ak4]), bG1 + kc + 16);
        async_wait_le3();
      } else {
        async_wait_0();
      }
      __syncthreads();
#pragma unroll
      for (int k0 = 0; k0 < 16; k0 += 4) {
        v2f a0, a1, b0, b1;
        a0.x = As[cur][w_m + lr][k0 + khalf];
        a0.y = As[cur][w_m + lr][k0 + khalf + 1];
        a1.x = As[cur][w_m + 16 + lr][k0 + khalf];
        a1.y = As[cur][w_m + 16 + lr][k0 + khalf + 1];
        b0.x = Bs[cur][w_n + lr][k0 + khalf];
        b0.y = Bs[cur][w_n + lr][k0 + khalf + 1];
        b1.x = Bs[cur][w_n + 16 + lr][k0 + khalf];
        b1.y = Bs[cur][w_n + 16 + lr][k0 + khalf + 1];
        c00 = __builtin_amdgcn_wmma_f32_16x16x4_f32(false, a0, false, b0,
                                                    (short)0, c00, false, false);
        c01 = __builtin_amdgcn_wmma_f32_16x16x4_f32(false, a0, false, b1,
                                                    (short)0, c01, false, false);
        c10 = __builtin_amdgcn_wmma_f32_16x16x4_f32(false, a1, false, b0,
                                                    (short)0, c10, false, false);
        c11 = __builtin_amdgcn_wmma_f32_16x16x4_f32(false, a1, false, b1,
                                                    (short)0, c11, false, false);
      }
      __syncthreads();
      cur ^= 1;
    }
    // Stage C tile: (vgpr r, lane l) -> row r + hi8, col lr (ISA C layout).
#pragma unroll
    for (int r = 0; r < 8; ++r) {
      const int r0 = w_m + hi8 + r;
      const int r1 = r0 + 16;
      Cs[r0][w_n + lr]      = c00[r];
      Cs[r0][w_n + 16 + lr] = c01[r];
      Cs[r1][w_n + lr]      = c10[r];
      Cs[r1][w_n + 16 + lr] = c11[r];
    }
    __syncthreads();
    if (tid < 64) {                    // thread tid owns tile row tid
      for (int j = 0; j < 128; ++j) {
        const int n = n0 + j;
        const float s = (n < CAP) ? Cs[tid][j] : -__builtin_inff();
        top3_insert(s, n, v0, v1, v2, i0, i1, i2);
      }
    }
    __syncthreads();
  }
  if (tid < 64) {
    const size_t base = ((size_t)chunk * B_ROWS + (m0 + tid)) * 3;
    pv[base + 0] = v0; pv[base + 1] = v1; pv[base + 2] = v2;
    pi[base + 0] = i0; pi[base + 1] = i1; pi[base + 2] = i2;
  }
}

// ---------------------------------------------------------------------------
// 3) Merge 196 per-chunk top-3 candidates -> global top-3; aug = query + s
// ---------------------------------------------------------------------------
__global__ __launch_bounds__(256) void merge_aug_kernel(
    const float* __restrict__ pv, const int* __restrict__ pi,
    const float* __restrict__ query, float* __restrict__ aug) {
  const int b = blockIdx.x;
  const int tid = threadIdx.x;
  __shared__ float s_add;
  if (tid == 0) {
    float v0 = -__builtin_inff(), v1 = -__builtin_inff(), v2 = -__builtin_inff();
    int i0 = 0, i1 = 0, i2 = 0;
    for (int c = 0; c < NCHUNKS; ++c) {
      const size_t base = ((size_t)c * B_ROWS + b) * 3;
#pragma unroll
      for (int t = 0; t < 3; ++t)
        top3_insert(pv[base + t], pi[base + t], v0, v1, v2, i0, i1, i2);
    }
    s_add = (float)(i0 + i1 + i2) * 0.001f;
  }
  __syncthreads();
  const float add = s_add;
  const float* qrow = query + (size_t)b * DIM;
  float* arow = aug + (size_t)b * DIM;
  for (int d = tid; d < DIM; d += 256) arow[d] = qrow[d] + add;
}

// ---------------------------------------------------------------------------
// 4/5) C = act(A[M,K] @ B[K,N] + bias), B row-major [K,N], exact GELU option.
//      Block tile 64x128; B staged non-transposed [k][n] via async B128
//      (column frag reads are conflict-free: lanes hit consecutive banks).
//      Grid: x = mtile (fast => blocks sharing a w2 column-strip are adjacent
//      in launch order for L2 reuse), y = ntile.
// ---------------------------------------------------------------------------
template <bool GELU>
__global__ __launch_bounds__(256) void gemm_bias_kernel(
    const float* __restrict__ A, const float* __restrict__ Bm,
    const float* __restrict__ bias, float* __restrict__ Cg,
    int M, int N, int K) {
  const int mtile = blockIdx.x;
  const int ntile = blockIdx.y;
  const int tid  = threadIdx.x;
  const int wave = tid >> 5;
  const int lane = tid & 31;
  const int lr   = lane & 15;
  const int khalf = (lane >> 4) << 1;
  const int hi8  = (lane >> 4) << 3;
  const int w_m  = (wave >> 2) << 5;    // 0,32
  const int w_n  = (wave & 3) << 5;     // 0,32,64,96
  const int m0   = mtile * 64;
  const int n0   = ntile * 128;

  __shared__ float As[2][64][PADA];     // [m][k]
  __shared__ float Bs[2][16][132];      // [k][n], row = 528B (16B-aligned)

  const int arow = tid >> 2;            // 0..63
  const int ak4  = (tid & 3) << 2;      // 0,4,8,12
  const int bk0  = tid >> 5;            // 0..7  (k rows 0..7, +8 for 2nd op)
  const int bn4  = (lane) << 2;         // 0..124 (wait: lane is 0..31 -> *4)

  const float* aG  = A  + (size_t)(m0 + arow) * K + ak4;
  const float* bG0 = Bm + (size_t)bk0 * N + n0 + bn4;        // + kc*N per stage
  const float* bG1 = Bm + (size_t)(bk0 + 8) * N + n0 + bn4;  // + kc*N per stage

  // Prologue: stage kc=0 into buffer 0.
  async_copy_b128(lds_addr_of(&As[0][arow][ak4]), aG);
  async_copy_b128(lds_addr_of(&Bs[0][bk0][bn4]),     bG0);
  async_copy_b128(lds_addr_of(&Bs[0][bk0 + 8][bn4]), bG1);

  v8f c00 = {}, c01 = {}, c10 = {}, c11 = {};
  int cur = 0;
  for (int kc = 0; kc < K; kc += 16) {
    if (kc + 16 < K) {
      const int nb = cur ^ 1;
      async_copy_b128(lds_addr_of(&As[nb][arow][ak4]), aG + kc + 16);
      async_copy_b128(lds_addr_of(&Bs[nb][bk0][bn4]),     bG0 + (size_t)(kc + 16) * N);
      async_copy_b128(lds_addr_of(&Bs[nb][bk0 + 8][bn4]), bG1 + (size_t)(kc + 16) * N);
      async_wait_le3();
    } else {
      async_wait_0();
    }
    __syncthreads();
#pragma unroll
    for (int k0 = 0; k0 < 16; k0 += 4) {
      v2f a0, a1, b0, b1;
      a0.x = As[cur][w_m + lr][k0 + khalf];
      a0.y = As[cur][w_m + lr][k0 + khalf + 1];
      a1.x = As[cur][w_m + 16 + lr][k0 + khalf];
      a1.y = As[cur][w_m + 16 + lr][k0 + khalf + 1];
      b0.x = Bs[cur][k0 + khalf][w_n + lr];
      b0.y = Bs[cur][k0 + khalf + 1][w_n + lr];
      b1.x = Bs[cur][k0 + khalf][w_n + 16 + lr];
      b1.y = Bs[cur][k0 + khalf + 1][w_n + 16 + lr];
      c00 = __builtin_amdgcn_wmma_f32_16x16x4_f32(false, a0, false, b0,
                                                  (short)0, c00, false, false);
      c01 = __builtin_amdgcn_wmma_f32_16x16x4_f32(false, a0, false, b1,
                                                  (short)0, c01, false, false);
      c10 = __builtin_amdgcn_wmma_f32_16x16x4_f32(false, a1, false, b0,
                                                  (short)0, c10, false, false);
      c11 = __builtin_amdgcn_wmma_f32_16x16x4_f32(false, a1, false, b1,
                                                  (short)0, c11, false, false);
    }
    __syncthreads();
    cur ^= 1;
  }
#pragma unroll
  for (int r = 0; r < 8; ++r) {
    const int r0 = m0 + w_m + hi8 + r;
    const int r1 = r0 + 16;
    const int cA = n0 + w_n + lr;
    const int cB = cA + 16;
    float x00 = c00[r] + bias[cA];
    float x01 = c01[r] + bias[cB];
    float x10 = c10[r] + bias[cA];
    float x11 = c11[r] + bias[cB];
    if (GELU) {
      x00 = 0.5f * x00 * (1.0f + erff(x00 * 0.70710678118654752f));
      x01 = 0.5f * x01 * (1.0f + erff(x01 * 0.70710678118654752f));
      x10 = 0.5f * x10 * (1.0f + erff(x10 * 0.70710678118654752f));
      x11 = 0.5f * x11 * (1.0f + erff(x11 * 0.70710678118654752f));
    }
    Cg[(size_t)r0 * N + cA] = x00;
    Cg[(size_t)r0 * N + cB] = x01;
    Cg[(size_t)r1 * N + cA] = x10;
    Cg[(size_t)r1 * N + cB] = x11;
  }
}

// ---------------------------------------------------------------------------
// Host-side launch
// ---------------------------------------------------------------------------
extern "C" void kernel_launch(void* const* d_in, const int* in_sizes, int n_in,
                              void* d_out, int out_size, void* d_ws, size_t ws_size,
                              hipStream_t stream) {
  (void)in_sizes; (void)n_in; (void)out_size; (void)ws_size;
  const int*   ids   = (const int*)d_in[0];
  const float* embed = (const float*)d_in[1];
  const float* index = (const float*)d_in[2];
  const float* w1    = (const float*)d_in[3];
  const float* b1    = (const float*)d_in[4];
  const float* w2    = (const float*)d_in[5];
  const float* b2    = (const float*)d_in[6];
  float* out = (float*)d_out;

  // Workspace layout (16B-aligned): ~27 MB total
  float* query = (float*)d_ws;                              // 1024*768
  float* qn    = query + (size_t)B_ROWS * DIM;              // 1024*768
  float* aug   = qn    + (size_t)B_ROWS * DIM;              // 1024*768
  float* pv    = aug   + (size_t)B_ROWS * DIM;              // 196*1024*3
  int*   pi    = (int*)(pv + (size_t)NCHUNKS * B_ROWS * 3); // 196*1024*3
  float* h     = (float*)(pi + (size_t)NCHUNKS * B_ROWS * 3); // 1024*3072

  query_kernel<<<B_ROWS, 256, 0, stream>>>(ids, embed, query, qn);
  scores_topk_kernel<<<dim3(B_ROWS / 64, NCHUNKS), 256, 0, stream>>>(qn, index, pv, pi);
  merge_aug_kernel<<<B_ROWS, 256, 0, stream>>>(pv, pi, query, aug);
  gemm_bias_kernel<true><<<dim3(B_ROWS / 64, HID / 128), 256, 0, stream>>>(
      aug, w1, b1, h, B_ROWS, HID, DIM);
  gemm_bias_kernel<false><<<dim3(B_ROWS / 64, NVOCAB / 128), 256, 0, stream>>>(
      h, w2, b2, out, B_ROWS, NVOCAB, HID);
}